// SparseLinear_44332652430010
// MI455X (gfx1250) — compile-verified
//
#include <hip/hip_runtime.h>

// SparseLinear: out[b,g,v] = sum_c w[g,v,c] * x[b, row_ind[g,c]]
// B=64, G=20000, V=16, NNZ=32.  Memory-bound (~133MB @ 23.3TB/s ~= 5.7us);
// the 1.31 GFLOP runs on the matrix pipe via V_WMMA_F32_16X16X4_F32 (full
// fp32 -> same numerics as the f32 reference).
// v2: 32-bit output offsets (SGPR-base+VGPR-offset stores, no 64-bit addr
// math in the WMMA loop); NT hints on streaming w loads / out stores so the
// L2-resident xT gather set (5MB of 192MB) isn't rinsed.

typedef float v2f __attribute__((ext_vector_type(2)));
typedef float v8f __attribute__((ext_vector_type(8)));

// ---------------------------------------------------------------------------
// Kernel 1: transpose x (64, G) -> xT (G, 64) so the per-index gather becomes
// a contiguous 256B row read (xT stays resident in the 192MB L2).
// ---------------------------------------------------------------------------
__global__ void transpose_x_kernel(const float* __restrict__ x,
                                   float* __restrict__ xT, int G) {
  __shared__ float tile[64 * 65];            // 65-float stride: bank-conflict free
  const int gt = blockIdx.x * 64;
  const int tx = threadIdx.x & 63;
  const int ty = threadIdx.x >> 6;           // 0..3
#pragma unroll
  for (int i = 0; i < 16; ++i) {
    const int b = ty + i * 4;                // 0..63
    const int g = gt + tx;
    if (g < G) tile[tx * 65 + b] = x[(size_t)b * G + g];   // coalesced read
  }
  __syncthreads();
#pragma unroll
  for (int i = 0; i < 16; ++i) {
    const int grow = ty + i * 4;
    const int g = gt + grow;
    if (g < G) xT[(size_t)g * 64 + tx] = tile[grow * 65 + tx]; // coalesced write
  }
}

// ---------------------------------------------------------------------------
// Kernel 2: one wave32 per group g.
//   D(64x16) = A(64x32) x B(32x16), A[b][c] = x[b, idx[c]], B[c][v] = w[g,v,c]
//   4 M-tiles (b) x 8 K-steps of V_WMMA_F32_16X16X4_F32, N = v (16).
// N==v orientation makes D stores coalesced (innermost out dim is v).
// ---------------------------------------------------------------------------
template <bool XT_LAYOUT>
__global__ void __launch_bounds__(128)
sparse_linear_kernel(const float* __restrict__ xsrc,   // xT (G,64) or x (64,G)
                     const int* __restrict__ row_ind,  // (G,32)
                     const float* __restrict__ w,      // (G,16,32)
                     float* __restrict__ out,          // (64,G,16)
                     int G) {
  constexpr int WPB = 4;    // waves per block
  constexpr int LS  = 34;   // LDS row stride (floats): even (8B-aligned b64
                            // fragment loads) and conflict-free (34L mod 64
                            // distinct for L=0..15, all-even bank bases)
  __shared__ float Xs[WPB * 64 * LS];        // 34,816 B

  const int wave = threadIdx.x >> 5;
  const int lane = threadIdx.x & 31;
  int g = blockIdx.x * WPB + wave;
  if (g >= G) g = G - 1;                     // uniform per wave; EXEC stays all-1s

  float* X = &Xs[wave * 64 * LS];            // this wave's 64 x 32 gathered tile

  // ---- gather: lane c owns column c; write X[b][c] for b=0..63 ----
  const int c   = lane;
  const int idx = row_ind[(size_t)g * 32 + c];
  if (XT_LAYOUT) {
    const float4* src = (const float4*)(xsrc + (size_t)idx * 64); // 256B row
#pragma unroll
    for (int q = 0; q < 16; ++q) {
      float4 vv = src[q];
      X[(q * 4 + 0) * LS + c] = vv.x;
      X[(q * 4 + 1) * LS + c] = vv.y;
      X[(q * 4 + 2) * LS + c] = vv.z;
      X[(q * 4 + 3) * LS + c] = vv.w;
    }
  } else {                                   // fallback: strided gather from x
#pragma unroll
    for (int b = 0; b < 64; ++b)
      X[b * LS + c] = xsrc[(size_t)b * G + idx];
  }
  __syncthreads();                           // uniform; orders LDS scatter vs reads

  // ---- B fragments (Wt, 4x16 per K-step): contiguous float2 from w ----
  // f32 A/B frag layout (ISA 7.12.2): lanes 0-15 hold {K0,K1}, lanes 16-31 {K2,K3}
  const int n    = lane & 15;                // N index (= v)
  const int koff = (lane >> 4) << 1;         // 0 or 2
  const float* wg = w + (size_t)g * (16 * 32);
  v2f bfrag[8];
#pragma unroll
  for (int kk = 0; kk < 8; ++kk)             // streaming: NT, don't pollute L2
    bfrag[kk] = __builtin_nontemporal_load(
        (const v2f*)(wg + n * 32 + kk * 4 + koff));          // global_load_b64

  // ---- 4 M-tiles x 8 K-steps of WMMA ----
  v8f acc[4];
#pragma unroll
  for (int t = 0; t < 4; ++t) acc[t] = (v8f)0.0f;

#pragma unroll
  for (int t = 0; t < 4; ++t) {
    const int b = t * 16 + (lane & 15);      // A-frag M row for this lane
#pragma unroll
    for (int kk = 0; kk < 8; ++kk) {
      v2f afrag = *(const v2f*)&X[b * LS + kk * 4 + koff];   // ds_load_2addr_b64
      acc[t] = __builtin_amdgcn_wmma_f32_16x16x4_f32(
          /*neg_a=*/false, afrag, /*neg_b=*/false, bfrag[kk],
          /*c_mod=*/(short)0, acc[t], /*reuse_a=*/false, /*reuse_b=*/false);
    }
  }

  // ---- store D tiles: lane = N = v; VGPR r -> M = r (+8 for hi lanes) ----
  // 32-bit element offsets (max ~20.5M < 2^31): SGPR base + 32b VGPR offset,
  // no 64-bit address math in the epilogue.
  const unsigned v       = lane & 15;
  const unsigned mbase   = (lane >> 4) * 8;
  const unsigned bstride = (unsigned)G * 16u;          // elems between b rows
  const unsigned baseoff = (unsigned)g * 16u + v;
#pragma unroll
  for (int t = 0; t < 4; ++t) {
#pragma unroll
    for (int r = 0; r < 8; ++r) {
      const unsigned b = (unsigned)t * 16u + mbase + (unsigned)r;
      __builtin_nontemporal_store(acc[t][r], out + (baseoff + b * bstride));
    }
  }
}

// ---------------------------------------------------------------------------
extern "C" void kernel_launch(void* const* d_in, const int* in_sizes, int n_in,
                              void* d_out, int out_size, void* d_ws, size_t ws_size,
                              hipStream_t stream) {
  const float* x       = (const float*)d_in[0];   // (64, G) f32
  const int*   row_ind = (const int*)d_in[1];     // (G, 32) int
  const float* w       = (const float*)d_in[2];   // (G, 16, 32) f32
  float*       out     = (float*)d_out;           // (64, G, 16) f32

  const int G = in_sizes[0] / 64;                 // B = 64 fixed
  const int main_blocks = (G + 3) / 4;            // 4 waves (groups) per block

  const size_t xt_bytes = (size_t)G * 64 * sizeof(float);
  if (d_ws != nullptr && ws_size >= xt_bytes) {
    float* xT = (float*)d_ws;
    transpose_x_kernel<<<(G + 63) / 64, 256, 0, stream>>>(x, xT, G);
    sparse_linear_kernel<true><<<main_blocks, 128, 0, stream>>>(xT, row_ind, w, out, G);
  } else {
    sparse_linear_kernel<false><<<main_blocks, 128, 0, stream>>>(x, row_ind, w, out, G);
  }
}